// ProbaVEval_45440753991930
// MI455X (gfx1250) — compile-verified
//
#include <hip/hip_runtime.h>
#include <hip/hip_bf16.h>
#include <math.h>

// ProbaV cPSNR:  loss(u,v) = Var(ct - cp(u,v)) ; score = -10*log10(min loss)
// Single pass: per offset need S1 = sum(d), S2 = sum(d^2), d = T - P_shifted.

#define HDIM   384
#define CROPN  3
#define HW     378              // 384 - 6
#define ROWS   21               // rows of the 378-grid per workgroup
#define NCHUNK 18               // 21*18 = 378
#define NOFF   36
#define NACC   72               // 36 offsets * (S1,S2)
#define GSTRIDE 80              // padded per-group partial stride (5 x 16 WMMA cols)
#define NGROUPS (32 * NCHUNK)   // 576
#define NPIX_F 4572288.0f       // 32*378*378

typedef __attribute__((ext_vector_type(2))) float v2f;
typedef __attribute__((ext_vector_type(8))) float v8f;

// ---------------------------------------------------------------------------
// Kernel A: per (batch, row-chunk) workgroup computes 72 partial sums.
// Masked predict tile (27 rows x 384 cols) staged in LDS with halo; each pixel
// reads its masked target once and streams 36 shifted LDS reads.
// ---------------------------------------------------------------------------
__global__ __launch_bounds__(256)
void probav_partials_kernel(const float* __restrict__ predict,
                            const float* __restrict__ target,
                            const float* __restrict__ mask,
                            float* __restrict__ ws) {
    __shared__ float4 Ptile4[(ROWS + 6) * (HDIM / 4)];   // 27*96 float4 = 41472 B
    __shared__ float  wred[8 * GSTRIDE];                 // 8 waves * 80
    float* Ptile = (float*)Ptile4;

    const int b     = blockIdx.x;          // batch 0..31
    const int chunk = blockIdx.y;          // 0..17
    const int tid   = threadIdx.x;
    const int r0    = chunk * ROWS;        // first row of this chunk (378-grid)
    const int base  = b * HDIM * HDIM;

    // ---- stage predict*mask tile rows [r0, r0+26] (float4 vectorized) ----
    const float4* p4 = (const float4*)(predict + base);
    const float4* m4 = (const float4*)(mask + base);
    for (int idx = tid; idx < (ROWS + 6) * (HDIM / 4); idx += 256) {
        int gr = r0 + idx / (HDIM / 4);
        int c4 = idx - (idx / (HDIM / 4)) * (HDIM / 4);
        float4 pv = p4[gr * (HDIM / 4) + c4];
        float4 mv = m4[gr * (HDIM / 4) + c4];
        float4 t;
        t.x = pv.x * mv.x; t.y = pv.y * mv.y;
        t.z = pv.z * mv.z; t.w = pv.w * mv.w;
        Ptile4[idx] = t;
    }
    __syncthreads();

    // ---- accumulate 36 offsets: acc[2o]=sum d, acc[2o+1]=sum d^2 ----
    float acc[NACC];
#pragma unroll
    for (int k = 0; k < NACC; ++k) acc[k] = 0.0f;

    for (int pi = tid; pi < ROWS * HW; pi += 256) {
        int r = pi / HW;                   // local row in 378-grid chunk
        int c = pi - r * HW;               // col 0..377
        int gidx = base + (r0 + r + CROPN) * HDIM + (c + CROPN);
        float T = target[gidx] * mask[gidx];
        const float* prow = &Ptile[r * HDIM + c];  // (r+u)*384 + (c+v) offsets
#pragma unroll
        for (int u = 0; u < 6; ++u) {
#pragma unroll
            for (int v = 0; v < 6; ++v) {
                float P = prow[u * HDIM + v];
                float d = T - P;
                const int o = u * 6 + v;
                acc[2 * o]     += d;
                acc[2 * o + 1]  = fmaf(d, d, acc[2 * o + 1]);
            }
        }
    }

    // ---- deterministic reduction: wave32 shuffle tree ----
#pragma unroll
    for (int k = 0; k < NACC; ++k) {
        float v = acc[k];
        v += __shfl_down(v, 16, 32);
        v += __shfl_down(v, 8, 32);
        v += __shfl_down(v, 4, 32);
        v += __shfl_down(v, 2, 32);
        v += __shfl_down(v, 1, 32);
        acc[k] = v;
    }
    const int lane = tid & 31;
    const int wid  = tid >> 5;
    if (lane == 0) {
#pragma unroll
        for (int k = 0; k < NACC; ++k) wred[wid * GSTRIDE + k] = acc[k];
    }
    __syncthreads();

    // ---- combine 8 waves in fixed order; write padded 80-slot group record --
    if (tid < GSTRIDE) {
        float s = 0.0f;
        if (tid < NACC) {
#pragma unroll
            for (int w = 0; w < 8; ++w) s += wred[w * GSTRIDE + tid];
        }
        int group = b * NCHUNK + chunk;
        ws[group * GSTRIDE + tid] = s;     // pads (72..79) written as 0.0
    }
}

// ---------------------------------------------------------------------------
// Kernel B: one wave32. Sum 576 group records with V_WMMA_F32_16X16X4_F32:
// D = ones(16x4) x B(4x16) + C  => every D row = column sums of B, i.e. the
// sum over 4 groups for 16 output slots. K is fully summed, so the result is
// invariant to the exact K<->(vgpr,lane-half) mapping; only the documented
// C/D layout (N = lane&15) is needed to extract results.
// ---------------------------------------------------------------------------
__global__ __launch_bounds__(32)
void probav_finalize_kernel(const float* __restrict__ ws,
                            float* __restrict__ out) {
    __shared__ float sums[GSTRIDE];
    const int lane  = threadIdx.x;
    const int half  = lane >> 4;           // 0: lanes 0-15, 1: lanes 16-31
    const int col16 = lane & 15;

    for (int blk = 0; blk < 5; ++blk) {    // 5 x 16 = 80 columns
        const int col = blk * 16 + col16;
        v8f c = {};
        const v2f a = {1.0f, 1.0f};        // A = ones(16x4), layout-invariant
        for (int g0 = 0; g0 < NGROUPS; g0 += 4) {
            v2f bm;
            bm.x = ws[(g0 + 2 * half) * GSTRIDE + col];
            bm.y = ws[(g0 + 2 * half + 1) * GSTRIDE + col];
            // 8 args: (neg_a, A, neg_b, B, c_mod, C, reuse_a, reuse_b)
            c = __builtin_amdgcn_wmma_f32_16x16x4_f32(
                false, a, false, bm, (short)0, c, false, false);
        }
        // D VGPR0: lanes 0-15 hold (M=0, N=lane); all rows identical here.
        if (lane < 16) sums[col] = c[0];
    }
    __syncthreads();

    if (lane == 0) {
        const float invN = 1.0f / NPIX_F;
        float best = 3.402823466e38f;
        for (int o = 0; o < NOFF; ++o) {
            float s1 = sums[2 * o];
            float s2 = sums[2 * o + 1];
            float m  = s1 * invN;
            float loss = fmaf(-m, m, s2 * invN);   // E[d^2] - E[d]^2
            best = fminf(best, loss);
        }
        out[0] = -10.0f * log10f(best);
    }
}

// ---------------------------------------------------------------------------
extern "C" void kernel_launch(void* const* d_in, const int* in_sizes, int n_in,
                              void* d_out, int out_size, void* d_ws, size_t ws_size,
                              hipStream_t stream) {
    (void)in_sizes; (void)n_in; (void)out_size; (void)ws_size;
    const float* predict = (const float*)d_in[0];
    const float* target  = (const float*)d_in[1];
    const float* mask    = (const float*)d_in[2];
    float* out = (float*)d_out;
    float* ws  = (float*)d_ws;              // uses NGROUPS*GSTRIDE*4 = 184,320 B

    dim3 grid(32, NCHUNK);                  // 576 workgroups, fully overwrite ws
    probav_partials_kernel<<<grid, 256, 0, stream>>>(predict, target, mask, ws);
    probav_finalize_kernel<<<1, 32, 0, stream>>>(ws, out);
}